// Loss_MSE_DistSnake_noGrad_39444979647067
// MI455X (gfx1250) — compile-verified
//
#include <hip/hip_runtime.h>
#include <hip/hip_bf16.h>
#include <math.h>

// Problem constants (match reference)
#define B_ 2
#define H_ 256
#define W_ 256
#define N_ 128
#define NSTEPS_ 50
#define STEPSZ_ 0.1f
#define ALPHA_ 0.01f
#define BETA_ 0.01f
#define DMAX_ 15.0f
#define EXTGRADFAC_ 10.0f

typedef __attribute__((ext_vector_type(2))) float v2f;
typedef __attribute__((ext_vector_type(8))) float v8f;

// d_ws float layout:
//   A1pk @ 0    : [B][8 chunks][32 lanes] float2 — WMMA A operand for S matrix
//                 lane<16 : (dy, dx)      lane>=16 : (-(p0.d), 0)
//   A2pk @ 1024 : [B][8 chunks][32 lanes] float2 — WMMA A operand for E matrix
//                 lane<16 : (-2p0y,-2p0x) lane>=16 : (|p0|^2, 1)
//   dD   @ 2048 : [B][128]  |d|^2 + 1e-8
//   iD   @ 2304 : [B][128]  1 / dD
//   partials @ 2560 : 1024 per-block partial sums
#define SEG_A1   0
#define SEG_A2   1024
#define SEG_DD   2048
#define SEG_ID   2304
#define WS_PART  2560
#define NBLK_RENDER 1024

// ---------------------------------------------------------------------------
// Kernel 1: snake evolution.  grid = B_ blocks of N_ threads.
// One thread per node; LDS double-buffer for circular 2nd/4th differences.
// jnp.gradient of pred is evaluated on the fly (central diff, one-sided edges).
// Epilogue writes WMMA-layout-packed per-segment operand arrays.
// ---------------------------------------------------------------------------
__global__ __launch_bounds__(N_) void evolve_kernel(const float* __restrict__ pred,
                                                    const float* __restrict__ nodes,
                                                    float* __restrict__ segs) {
  const int b = blockIdx.x;
  const int i = threadIdx.x;
  const float* P = pred + b * H_ * W_;

  float y = nodes[(b * N_ + i) * 2 + 0];
  float x = nodes[(b * N_ + i) * 2 + 1];

  __shared__ float sy[N_], sx[N_], t2y[N_], t2x[N_];

  const int ip = (i + 1) & (N_ - 1);
  const int im = (i + N_ - 1) & (N_ - 1);
  const float cliphi = (float)(H_ - 1.001);  // bilerp clip bound (254.999)

  auto grad_y = [&](int r, int c) -> float {
    if (r == 0)      return P[W_ + c] - P[c];
    if (r == H_ - 1) return P[(H_ - 1) * W_ + c] - P[(H_ - 2) * W_ + c];
    return 0.5f * (P[(r + 1) * W_ + c] - P[(r - 1) * W_ + c]);
  };
  auto grad_x = [&](int r, int c) -> float {
    if (c == 0)      return P[r * W_ + 1] - P[r * W_];
    if (c == W_ - 1) return P[r * W_ + W_ - 1] - P[r * W_ + W_ - 2];
    return 0.5f * (P[r * W_ + c + 1] - P[r * W_ + c - 1]);
  };

  for (int step = 0; step < NSTEPS_; ++step) {
    sy[i] = y; sx[i] = x;
    __syncthreads();
    float d2y = sy[ip] + sy[im] - 2.0f * y;
    float d2x = sx[ip] + sx[im] - 2.0f * x;
    t2y[i] = d2y; t2x[i] = d2x;
    __syncthreads();
    float d4y = t2y[ip] + t2y[im] - 2.0f * d2y;
    float d4x = t2x[ip] + t2x[im] - 2.0f * d2x;

    // bilinear sample of (gy, gx) at (y, x)
    float yc = fminf(fmaxf(y, 0.0f), cliphi);
    float xc = fminf(fmaxf(x, 0.0f), cliphi);
    int y0 = (int)floorf(yc);
    int x0 = (int)floorf(xc);
    int y1 = y0 + 1, x1 = x0 + 1;
    float wy = yc - (float)y0;
    float wx = xc - (float)x0;
    float w00 = (1.0f - wy) * (1.0f - wx);
    float w01 = (1.0f - wy) * wx;
    float w10 = wy * (1.0f - wx);
    float w11 = wy * wx;

    float gyv = grad_y(y0, x0) * w00 + grad_y(y0, x1) * w01 +
                grad_y(y1, x0) * w10 + grad_y(y1, x1) * w11;
    float gxv = grad_x(y0, x0) * w00 + grad_x(y0, x1) * w01 +
                grad_x(y1, x0) * w10 + grad_x(y1, x1) * w11;

    y = y + STEPSZ_ * (ALPHA_ * d2y - BETA_ * d4y - EXTGRADFAC_ * gyv);
    x = x + STEPSZ_ * (ALPHA_ * d2x - BETA_ * d4x - EXTGRADFAC_ * gxv);
    y = fminf(fmaxf(y, 0.0f), (float)(H_ - 1));
    x = fminf(fmaxf(x, 0.0f), (float)(W_ - 1));
  }

  // epilogue: per-segment parameters (segment i = node i -> node i+1 mod N),
  // packed directly in WMMA A-operand lane layout.
  sy[i] = y; sx[i] = x;
  __syncthreads();
  float p1y = sy[ip], p1x = sx[ip];
  float dyv = p1y - y;
  float dxv = p1x - x;
  float ddv = dyv * dyv + dxv * dxv + 1e-8f;
  float ct  = -(y * dyv + x * dxv);
  float psq = y * y + x * x;

  const int c = i >> 4;       // segment chunk (0..7)
  const int n = i & 15;       // position within chunk
  float2* A1pk = (float2*)(segs + SEG_A1);
  float2* A2pk = (float2*)(segs + SEG_A2);
  const int base = (b * 8 + c) * 32;
  A1pk[base + n]      = make_float2(dyv, dxv);
  A1pk[base + n + 16] = make_float2(ct, 0.0f);
  A2pk[base + n]      = make_float2(-2.0f * y, -2.0f * x);
  A2pk[base + n + 16] = make_float2(psq, 1.0f);

  const int o = b * N_ + i;
  segs[SEG_DD + o] = ddv;
  segs[SEG_ID + o] = 1.0f / ddv;
}

// ---------------------------------------------------------------------------
// Kernel 2: render distance map + squared-error partial sums.
// Each wave owns 16 consecutive pixels (one row) x all 128 segments.
// Per chunk of 16 segments: two V_WMMA_F32_16X16X4_F32 ops compute
//   S[m][n] = (q_n - p0_m) . d_m
//   E[m][n] = |q_n - p0_m|^2
// VALU epilogue: t = clamp(S/dd), dist^2 = E - 2tS + t^2 dd, running min.
// M = segments, N = pixels -> per-pixel min is 7 VALU mins + one shfl_xor(16).
// Chunk loop fully unrolled: 16 static WMMAs, branchless pre-packed A loads.
// ---------------------------------------------------------------------------
__global__ __launch_bounds__(256) void render_loss_kernel(const float* __restrict__ pred,
                                                          const float* __restrict__ segs,
                                                          float* __restrict__ partials) {
  const int tid   = threadIdx.x;
  const int wslot = tid >> 5;                 // wave within block (0..7)
  const int w     = blockIdx.x * 8 + wslot;   // global wave id (0..8191)
  const int b     = w >> 12;                  // batch
  const int row   = (w >> 4) & (H_ - 1);      // pixel row
  const int xb    = (w & 15) << 4;            // pixel col base
  const int lane  = tid & 31;
  const int n     = lane & 15;
  const bool lo   = lane < 16;

  const float qy = (float)row;
  const float qx = (float)(xb + n);

  // B operands (constant over segment chunks); built once, outside the
  // WMMA region, so EXEC is all-ones at every WMMA.
  v2f B1; B1[0] = lo ? qy : 1.0f; B1[1] = lo ? qx : 0.0f;
  v2f B2; B2[0] = B1[0];          B2[1] = lo ? qx : fmaf(qy, qy, qx * qx);

  const float2* A1pk = (const float2*)(segs + SEG_A1);
  const float2* A2pk = (const float2*)(segs + SEG_A2);
  const float*  dD   = segs + SEG_DD;
  const float*  iD   = segs + SEG_ID;

  float md[8];
#pragma unroll
  for (int r = 0; r < 8; ++r) md[r] = 3.0e38f;

  const int hiOff = lo ? 0 : 8;  // segment row offset for lanes 16..31

#pragma unroll
  for (int c = 0; c < N_ / 16; ++c) {
    // branchless, coalesced A-operand loads (pre-packed in WMMA lane layout)
    const float2 a1 = A1pk[(b * 8 + c) * 32 + lane];
    const float2 a2 = A2pk[(b * 8 + c) * 32 + lane];
    v2f A1; A1[0] = a1.x; A1[1] = a1.y;
    v2f A2; A2[0] = a2.x; A2[1] = a2.y;

    v8f z = {0.f, 0.f, 0.f, 0.f, 0.f, 0.f, 0.f, 0.f};
    v8f S = __builtin_amdgcn_wmma_f32_16x16x4_f32(false, A1, false, B1,
                                                  (short)0, z, false, false);
    v8f E = __builtin_amdgcn_wmma_f32_16x16x4_f32(false, A2, false, B2,
                                                  (short)0, z, false, false);

    const int segE = b * N_ + c * 16 + hiOff;  // segment for C-reg r is segE + r
#pragma unroll
    for (int r = 0; r < 8; ++r) {
      float s   = S[r];
      float dd  = dD[segE + r];
      float inv = iD[segE + r];
      float t   = fminf(fmaxf(s * inv, 0.0f), 1.0f);
      float d2  = fmaf(t, fmaf(t, dd, -2.0f * s), E[r]);
      md[r] = fminf(md[r], d2);
    }
  }

  // per-pixel min over segments: 7 VALU mins + cross-half combine
  float m = md[0];
#pragma unroll
  for (int r = 1; r < 8; ++r) m = fminf(m, md[r]);
  m = fminf(m, __shfl_xor(m, 16));

  float dm  = fminf(sqrtf(fmaxf(m, 0.0f) + 1e-12f), DMAX_);
  float pv  = pred[b * H_ * W_ + row * W_ + xb + n];
  float err = pv - dm;
  float sq  = lo ? err * err : 0.0f;  // lanes 16..31 hold duplicate pixels

  // deterministic wave reduction (fixed butterfly order)
#pragma unroll
  for (int off = 16; off > 0; off >>= 1) sq += __shfl_xor(sq, off);

  __shared__ float wsum[8];
  if (lane == 0) wsum[wslot] = sq;
  __syncthreads();
  if (tid == 0) {
    float t = 0.0f;
#pragma unroll
    for (int k = 0; k < 8; ++k) t += wsum[k];
    partials[blockIdx.x] = t;
  }
}

// ---------------------------------------------------------------------------
// Kernel 3: deterministic final reduction -> loss scalar.
// ---------------------------------------------------------------------------
__global__ __launch_bounds__(256) void finalize_kernel(const float* __restrict__ partials,
                                                       float* __restrict__ out) {
  __shared__ float acc[256];
  const int t = threadIdx.x;
  float s = 0.0f;
  for (int i = t; i < NBLK_RENDER; i += 256) s += partials[i];
  acc[t] = s;
  __syncthreads();
  if (t == 0) {
    float tot = 0.0f;
    for (int k = 0; k < 256; ++k) tot += acc[k];
    out[0] = tot * (1.0f / (float)(B_ * H_ * W_));
  }
}

extern "C" void kernel_launch(void* const* d_in, const int* in_sizes, int n_in,
                              void* d_out, int out_size, void* d_ws, size_t ws_size,
                              hipStream_t stream) {
  const float* pred  = (const float*)d_in[0];  // (B,1,H,W) float32, 131072
  const float* nodes = (const float*)d_in[1];  // (B,N,2)   float32, 512
  float* ws       = (float*)d_ws;
  float* segs     = ws;             // packed operand + segment arrays
  float* partials = ws + WS_PART;   // 1024 per-block partials
  float* out      = (float*)d_out;  // scalar loss

  evolve_kernel<<<B_, N_, 0, stream>>>(pred, nodes, segs);
  render_loss_kernel<<<NBLK_RENDER, 256, 0, stream>>>(pred, segs, partials);
  finalize_kernel<<<1, 256, 0, stream>>>(partials, out);
}